// MinkowskiInstanceNorm_13881334301293
// MI455X (gfx1250) — compile-verified
//
#include <hip/hip_runtime.h>

typedef __attribute__((ext_vector_type(2))) float v2f;
typedef __attribute__((ext_vector_type(8))) float v8f;

#define NSEG 16
#define CH   64
#define STATS_F (NSEG * CH)      // 1024 floats per statistic
#define P1_BLOCKS 1024
#define EPSV 1e-8f

// ---------------------------------------------------------------------------
// Kernel 0: zero the sum / sumsq accumulators (must run every call: graph
// replays accumulate atomically into this region).
// ---------------------------------------------------------------------------
__global__ void zero_ws(float* __restrict__ ws) {
    int i = blockIdx.x * blockDim.x + threadIdx.x;
    if (i < 2 * STATS_F) ws[i] = 0.0f;
}

// ---------------------------------------------------------------------------
// Kernel 1: segment sum + sum-of-squares via V_WMMA_F32_16X16X4_F32.
//   D[seg, ch] += onehot(batch_id)[seg, k] * feats[k, ch]
// A (16x4, fp32): lanes 0-15 hold K=0(v0),K=1(v1); lanes 16-31 hold K=2,K=3.
// B (4x16, fp32): v0 = rows K=0 (lanes 0-15) / K=2 (lanes 16-31); v1 = K=1/K=3.
// C/D (16x16, fp32): lane<16 -> M=v, lane>=16 -> M=v+8; N = lane&15.
// ---------------------------------------------------------------------------
__global__ void __launch_bounds__(256)
pass1_wmma(const float* __restrict__ feats, const int* __restrict__ ids,
           float* __restrict__ ws_sum, float* __restrict__ ws_sq, long N) {
    __shared__ float lds[2 * STATS_F];     // [sum(16x64) | sumsq(16x64)] = 8KB
    const int tid  = threadIdx.x;
    const int wave = tid >> 5;
    const int lane = tid & 31;
    const int half = lane >> 4;            // which K-pair / which M-half
    const int m    = lane & 15;            // A: row M ; B: column N

    for (int i = tid; i < 2 * STATS_F; i += 256) lds[i] = 0.0f;
    __syncthreads();

    v8f accS[4], accQ[4];
    #pragma unroll
    for (int t = 0; t < 4; ++t) {
        #pragma unroll
        for (int v = 0; v < 8; ++v) { accS[t][v] = 0.0f; accQ[t][v] = 0.0f; }
    }

    const long ngroups = N >> 2;                                   // 4 rows/group
    const long gpb     = (ngroups + gridDim.x - 1) / gridDim.x;
    const long base    = (long)blockIdx.x * gpb;
    const long gend    = (base + gpb < ngroups) ? (base + gpb) : ngroups;

    for (long g = base + wave; g < gend; g += 8) {                 // wave-uniform
        const long row0 = g << 2;
        const long rA   = row0 + 2 * half;                         // K = 2*half
        const long rB   = rA + 1;                                  // K = 2*half+1

        const int idA = ids[rA];
        const int idB = ids[rB];
        v2f A;
        A[0] = (idA == m) ? 1.0f : 0.0f;
        A[1] = (idB == m) ? 1.0f : 0.0f;

        const float* pA = feats + rA * CH + m;
        const float* pB = feats + rB * CH + m;
        __builtin_prefetch(pA + 32 * CH, 0, 0);                    // next iter (+32 rows)
        __builtin_prefetch(pB + 32 * CH, 0, 0);

        #pragma unroll
        for (int t = 0; t < 4; ++t) {                              // 4 channel tiles
            v2f B, Q;
            B[0] = pA[t * 16];
            B[1] = pB[t * 16];
            Q[0] = B[0] * B[0];
            Q[1] = B[1] * B[1];
            accS[t] = __builtin_amdgcn_wmma_f32_16x16x4_f32(
                false, A, false, B, (short)0, accS[t], false, false);
            accQ[t] = __builtin_amdgcn_wmma_f32_16x16x4_f32(
                false, A, false, Q, (short)0, accQ[t], false, false);
        }
    }

    // Flush wave-private tiles into the block-shared accumulator (ds_add_f32).
    #pragma unroll
    for (int t = 0; t < 4; ++t) {
        const int chan = t * 16 + m;
        #pragma unroll
        for (int v = 0; v < 8; ++v) {
            const int seg = v + 8 * half;
            atomicAdd(&lds[seg * CH + chan],           accS[t][v]);
            atomicAdd(&lds[STATS_F + seg * CH + chan], accQ[t][v]);
        }
    }
    __syncthreads();

    for (int i = tid; i < STATS_F; i += 256) {
        unsafeAtomicAdd(&ws_sum[i], lds[i]);                       // global_atomic_add_f32
        unsafeAtomicAdd(&ws_sq[i],  lds[STATS_F + i]);
    }
}

// ---------------------------------------------------------------------------
// Kernel 2: per-(segment,channel) fused scale/shift.
//   counts from binary search on sorted batch_ids (16 segments).
//   a = weight/sqrt(var+eps), b = bias - mean*a  ->  out = x*a + b
// ---------------------------------------------------------------------------
__global__ void __launch_bounds__(1024)
stats_kernel(const float* __restrict__ ws_sum, const float* __restrict__ ws_sq,
             const int* __restrict__ ids, const float* __restrict__ weight,
             const float* __restrict__ bias, float* __restrict__ ws_a,
             float* __restrict__ ws_b, long N) {
    const int t = threadIdx.x;
    if (t >= STATS_F) return;
    const int s = t >> 6;
    const int c = t & 63;

    auto lower_bound = [&](int key) -> long {
        long lo = 0, hi = N;
        while (lo < hi) {
            long mid = (lo + hi) >> 1;
            if (ids[mid] < key) lo = mid + 1; else hi = mid;
        }
        return lo;
    };
    const long b0 = lower_bound(s);
    const long b1 = lower_bound(s + 1);
    float cnt = (float)(b1 - b0);
    cnt = fmaxf(cnt, 1.0f);

    const float mean = ws_sum[t] / cnt;
    float var = ws_sq[t] / cnt - mean * mean;
    var = fmaxf(var, 0.0f);
    const float inv = 1.0f / sqrtf(var + EPSV);
    const float a   = inv * weight[c];
    ws_a[t] = a;
    ws_b[t] = bias[c] - mean * a;
}

// ---------------------------------------------------------------------------
// Kernel 3: streaming apply, float4 vectorized (global_load/store_b128).
// 16 threads per row (16 x float4 = 64 channels); a/b tables staged in LDS.
// ---------------------------------------------------------------------------
__global__ void __launch_bounds__(256)
pass2_apply(const float* __restrict__ feats, const int* __restrict__ ids,
            const float* __restrict__ ws_a, const float* __restrict__ ws_b,
            float* __restrict__ out, long N) {
    __shared__ float sa[STATS_F];
    __shared__ float sb[STATS_F];
    for (int i = threadIdx.x; i < STATS_F; i += 256) {
        sa[i] = ws_a[i];
        sb[i] = ws_b[i];
    }
    __syncthreads();

    const long row = (long)blockIdx.x * 16 + (threadIdx.x >> 4);
    if (row >= N) return;
    const int c4 = (threadIdx.x & 15) * 4;

    const int  s   = ids[row];
    const int  off = s * CH + c4;
    const float4 x = *reinterpret_cast<const float4*>(feats + row * CH + c4);
    const float4 a = *reinterpret_cast<const float4*>(&sa[off]);
    const float4 b = *reinterpret_cast<const float4*>(&sb[off]);

    float4 r;
    r.x = fmaf(x.x, a.x, b.x);
    r.y = fmaf(x.y, a.y, b.y);
    r.z = fmaf(x.z, a.z, b.z);
    r.w = fmaf(x.w, a.w, b.w);
    *reinterpret_cast<float4*>(out + row * CH + c4) = r;
}

// ---------------------------------------------------------------------------
extern "C" void kernel_launch(void* const* d_in, const int* in_sizes, int n_in,
                              void* d_out, int out_size, void* d_ws, size_t ws_size,
                              hipStream_t stream) {
    const float* feats  = (const float*)d_in[0];
    const int*   ids    = (const int*)d_in[1];
    const float* weight = (const float*)d_in[2];
    const float* bias   = (const float*)d_in[3];
    float*       out    = (float*)d_out;
    const long   N      = (long)in_sizes[1];

    float* ws     = (float*)d_ws;          // needs 4*1024 floats = 16 KB
    float* ws_sum = ws;
    float* ws_sq  = ws + STATS_F;
    float* ws_a   = ws + 2 * STATS_F;
    float* ws_b   = ws + 3 * STATS_F;

    zero_ws<<<dim3(8), dim3(256), 0, stream>>>(ws);
    pass1_wmma<<<dim3(P1_BLOCKS), dim3(256), 0, stream>>>(feats, ids, ws_sum, ws_sq, N);
    stats_kernel<<<dim3(1), dim3(1024), 0, stream>>>(ws_sum, ws_sq, ids, weight, bias,
                                                     ws_a, ws_b, N);
    const unsigned nblk = (unsigned)((N + 15) / 16);
    pass2_apply<<<dim3(nblk), dim3(256), 0, stream>>>(feats, ids, ws_a, ws_b, out, N);
}